// SpanEmbedder_22548578304682
// MI455X (gfx1250) — compile-verified
//
#include <hip/hip_runtime.h>
#include <hip/hip_bf16.h>

// ---------------------------------------------------------------------------
// SpanEmbedder forward, MI455X (gfx1250, wave32).
//   f32 WMMA 16x16x4 for all GEMMs + double-buffered Tensor Data Mover (TDM)
//   pipeline for global->LDS staging with hardware padding (bank-conflict-
//   free WMMA operand reads). DMA of chunk k+1 overlaps WMMA of chunk k.
//   B=16, S=128, T=1024, M=16, H=1024, HID=1024 ; out dim D = 3094
// Workspace: cont (B*S*M*H f32 = 128MB) + scores (B*S*M f32).
// ---------------------------------------------------------------------------

#define B_   16
#define S_   128
#define T_   1024
#define M_   16
#define H_   1024
#define HID_ 1024
#define D_   3094

typedef float        v2f __attribute__((ext_vector_type(2)));
typedef float        v8f __attribute__((ext_vector_type(8)));
typedef unsigned int v4u __attribute__((ext_vector_type(4)));
typedef int          v4i __attribute__((ext_vector_type(4)));
typedef int          v8i __attribute__((ext_vector_type(8)));

// ---------------------------------------------------------------------------
// TDM 2D tile load: global (row-major, row stride in elements) -> LDS with
// padding of (pad_amount_code+1) DWORDs every 2^(pad_interval_code+1) DWORDs.
// D# packing per CDNA5 ISA §8.3/§8.4.
// ---------------------------------------------------------------------------
__device__ __forceinline__ void tdm_load_2d(unsigned lds_off, const void* gptr,
                                            int tile_w, int tile_h,
                                            long long row_stride_elts,
                                            int pad_interval_code,
                                            int pad_amount_code)
{
    unsigned long long ga = (unsigned long long)(uintptr_t)gptr;
    v4u g0;
    g0[0] = 1u;                                        // count=1, user mode
    g0[1] = lds_off;                                   // lds_addr (bytes)
    g0[2] = (unsigned)(ga & 0xffffffffull);            // global_addr[31:0]
    g0[3] = (unsigned)((ga >> 32) & 0x1ffffffull)      // global_addr[56:32]
          | (2u << 30);                                // type = 2 ("image")
    unsigned td0 = (unsigned)tile_w;                   // tensor_dim0 (elts)
    unsigned td1 = (unsigned)tile_h;                   // tensor_dim1
    unsigned long long st0 = (unsigned long long)row_stride_elts;
    v8i g1;
    g1[0] = (int)((2u << 16)                           // data_size = 4 bytes
                | (1u << 20)                           // pad_enable
                | ((unsigned)pad_interval_code << 22)
                | ((unsigned)pad_amount_code << 25));
    g1[1] = (int)((td0 & 0xffffu) << 16);              // tensor_dim0[15:0]
    g1[2] = (int)(((td0 >> 16) & 0xffffu) | ((td1 & 0xffffu) << 16));
    g1[3] = (int)(((td1 >> 16) & 0xffffu) | (((unsigned)tile_w & 0xffffu) << 16)); // tile_dim0
    g1[4] = (int)((unsigned)tile_h & 0xffffu);         // tile_dim1 (tile_dim2=0)
    g1[5] = (int)(st0 & 0xffffffffull);                // tensor_dim0_stride lo
    g1[6] = (int)((st0 >> 32) & 0xffffull);            // stride hi, dim1_stride=0
    g1[7] = 0;
    v4i z4 = {0, 0, 0, 0};
#if __clang_major__ >= 23
    v8i z8 = {0, 0, 0, 0, 0, 0, 0, 0};
    __builtin_amdgcn_tensor_load_to_lds(g0, g1, z4, z4, z8, 0);
#else
    __builtin_amdgcn_tensor_load_to_lds(g0, g1, z4, z4, 0);
#endif
}

__device__ __forceinline__ unsigned lds_offset_of(const void* p)
{
    // Low 32 bits of a flat shared pointer == LDS byte offset on AMDGPU.
    return (unsigned)(uintptr_t)p;
}

// ---------------------------------------------------------------------------
// Kernel A: generic batched f32 GEMM  C[b] = A[b] (rows x K) * Bm[b] (K x N)
// Block = 256 threads = 8 waves arranged 4(row) x 2(col) -> 64x32 C tile.
// Double-buffered TDM pipeline over 32-wide K chunks:
//   issue chunk k+1 into buf^1, s_wait_tensorcnt(2) for chunk k's two loads,
//   barrier, 8x v_wmma_f32_16x16x4_f32 from LDS buf, barrier, swap.
// ---------------------------------------------------------------------------
__global__ __launch_bounds__(256)
void se_wmma_gemm_f32(const float* __restrict__ A, long long sAb, int lda,
                      const float* __restrict__ Bm, long long sBb, int ldb,
                      float* __restrict__ C, long long sCb, int ldc, int K)
{
    const int b = blockIdx.z;
    A  += (long long)b * sAb;
    Bm += (long long)b * sBb;
    C  += (long long)b * sCb;

    const int tid  = threadIdx.x;
    const int wave = tid >> 5;
    const int lane = tid & 31;
    const int wr   = wave >> 1;          // 0..3 : 16-row band within block
    const int wc   = wave & 1;           // 0..1 : 16-col band within block
    const int l15  = lane & 15;
    const int ksel = (lane < 16) ? 0 : 2;     // K half per A/B f32 layout
    const int rowBlk = blockIdx.y * 64;
    const int colBlk = blockIdx.x * 32;

    __shared__ float As[2][64][33];   // TDM pads 1 DWORD per 32 -> stride 33
    __shared__ float Bs[2][32][33];

    const unsigned asOff0 = lds_offset_of(&As[0][0][0]);
    const unsigned asOff1 = lds_offset_of(&As[1][0][0]);
    const unsigned bsOff0 = lds_offset_of(&Bs[0][0][0]);
    const unsigned bsOff1 = lds_offset_of(&Bs[1][0][0]);

    v8f acc = {};

    // prologue: chunk 0 into buffer 0
    if (wave == 0) {
        tdm_load_2d(asOff0, A + (long long)rowBlk * lda, 32, 64,
                    (long long)lda, /*interval 32 dw*/4, /*pad 1 dw*/0);
        tdm_load_2d(bsOff0, Bm + colBlk, 32, 32, (long long)ldb, 4, 0);
    }

    int buf = 0;
    for (int kk = 0; kk < K; kk += 32) {
        if (wave == 0) {
            if (kk + 32 < K) {
                // issue chunk k+1 into the other buffer, then wait only for
                // chunk k's two loads (TDM completes in order per wave).
                const unsigned ao = buf ? asOff0 : asOff1;
                const unsigned bo = buf ? bsOff0 : bsOff1;
                tdm_load_2d(ao, A + (long long)rowBlk * lda + (kk + 32),
                            32, 64, (long long)lda, 4, 0);
                tdm_load_2d(bo, Bm + (long long)(kk + 32) * ldb + colBlk,
                            32, 32, (long long)ldb, 4, 0);
                __builtin_amdgcn_s_wait_tensorcnt(2);
            } else {
                __builtin_amdgcn_s_wait_tensorcnt(0);
            }
        }
        __syncthreads();

        #pragma unroll
        for (int k4 = 0; k4 < 32; k4 += 4) {
            v2f a, bf;
            const int ar = wr * 16 + l15;           // A row = C row (M)
            a.x = As[buf][ar][k4 + ksel];
            a.y = As[buf][ar][k4 + ksel + 1];
            const int bc = wc * 16 + l15;           // B col = C col (N)
            bf.x = Bs[buf][k4 + ksel][bc];
            bf.y = Bs[buf][k4 + ksel + 1][bc];
            acc = __builtin_amdgcn_wmma_f32_16x16x4_f32(
                      false, a, false, bf, (short)0, acc, false, false);
        }
        __syncthreads();   // all waves done reading buf before it is refilled
        buf ^= 1;
    }

    // C/D layout: VGPR g -> row g (lanes 0-15) / row 8+g (lanes 16-31)
    #pragma unroll
    for (int g = 0; g < 8; ++g) {
        int row = rowBlk + wr * 16 + g + ((lane < 16) ? 0 : 8);
        int col = colBlk + wc * 16 + l15;
        C[(long long)row * ldc + col] = acc[g];
    }
}

// ---------------------------------------------------------------------------
// Kernel B: fused attention-MLP scores.
// One block (4 waves) per span n. cont block (16 x 1024 f32) staged once via
// TDM with 1-DWORD pad every 256 -> row stride 1028 (banks 4 apart).
// Each wave sweeps 16 hid-tiles with WMMA; epilogue fuses +b1, ReLU, *W2.
// ---------------------------------------------------------------------------
#define BROW_ 1028                       // padded row stride (floats)
__device__ __forceinline__ int asm_idx(int r, int c)
{
    return r * BROW_ + c + (c >> 8);     // TDM pad: +1 DWORD per 256 DWORDs
}

__global__ __launch_bounds__(128)
void se_score_kernel(const float* __restrict__ cont,
                     const float* __restrict__ W1,
                     const float* __restrict__ b1,
                     const float* __restrict__ W2,
                     const float* __restrict__ b2,
                     float* __restrict__ scores)
{
    const int n    = blockIdx.x;
    const int tid  = threadIdx.x;
    const int wave = tid >> 5;
    const int lane = tid & 31;
    const int l15  = lane & 15;
    const int ksel = (lane < 16) ? 0 : 2;

    __shared__ float Asm[M_ * BROW_];    // 16 rows, padded
    __shared__ float sred[16];

    const float* crow = cont + (long long)n * M_ * H_;
    if (wave == 0) {
        // whole 16x1024 block, rows contiguous (stride 1024 elements)
        tdm_load_2d(lds_offset_of(&Asm[0]), crow, H_, M_, (long long)H_,
                    /*interval 256 dw*/7, /*pad 1 dw*/0);
        __builtin_amdgcn_s_wait_tensorcnt(0);
    }
    if (tid < 16) sred[tid] = 0.0f;
    __syncthreads();

    float pacc[8];
    #pragma unroll
    for (int g = 0; g < 8; ++g) pacc[g] = 0.0f;

    for (int j = wave; j < HID_ / 16; j += 4) {
        const int colb = j * 16;
        v8f c = {};
        for (int kk = 0; kk < H_; kk += 4) {
            v2f a, bf;
            const int ai = asm_idx(l15, kk + ksel);   // (even,odd) pair never
            a.x = Asm[ai];                            // straddles a pad point
            a.y = Asm[ai + 1];
            bf.x = W1[(long long)(kk + ksel)     * HID_ + colb + l15];
            bf.y = W1[(long long)(kk + ksel + 1) * HID_ + colb + l15];
            c = __builtin_amdgcn_wmma_f32_16x16x4_f32(
                    false, a, false, bf, (short)0, c, false, false);
        }
        #pragma unroll
        for (int g = 0; g < 8; ++g) {
            int col = colb + l15;
            float h = c[g] + b1[col];
            h = h > 0.0f ? h : 0.0f;              // ReLU
            pacc[g] += h * W2[col];               // fuse * W2 (HID x 1)
        }
    }

    // reduce the 16 columns: lanes 0-15 hold rows g, lanes 16-31 rows 8+g
    #pragma unroll
    for (int g = 0; g < 8; ++g) {
        float v = pacc[g];
        v += __shfl_xor(v, 1, 32);
        v += __shfl_xor(v, 2, 32);
        v += __shfl_xor(v, 4, 32);
        v += __shfl_xor(v, 8, 32);
        pacc[g] = v;
    }
    if (l15 == 0) {                               // lane 0 and lane 16
        int rbase = (lane < 16) ? 0 : 8;
        #pragma unroll
        for (int g = 0; g < 8; ++g)
            atomicAdd(&sred[rbase + g], pacc[g]); // ds_add_f32
    }
    __syncthreads();
    if (tid < 16) scores[n * M_ + tid] = sred[tid] + b2[0];
}

// ---------------------------------------------------------------------------
// Kernel C: masked softmax (exact reference semantics), weighted sum of cont,
// linguistic columns, clamped width embedding gather.
// ---------------------------------------------------------------------------
__global__ __launch_bounds__(256)
void se_finalize_kernel(const float* __restrict__ cont,
                        const float* __restrict__ scores,
                        const float* __restrict__ ling,
                        const float* __restrict__ wtab,
                        const int*   __restrict__ width,
                        float* __restrict__ out)
{
    const int n   = blockIdx.x;
    const int tid = threadIdx.x;

    __shared__ float msc[16];
    if (tid < 16) {
        int w = width[n];
        int wmax = (w > 1) ? w : 1;
        float s = scores[n * M_ + tid];
        s *= (tid < wmax) ? 1.0f : 0.0f;          // scores * mask
        msc[tid] = (s != 0.0f) ? s : -9e9f;       // where(s!=0, s, -9e9)
    }
    __syncthreads();

    // softmax over M=16 (computed redundantly per thread, cheap)
    float mx = msc[0];
    #pragma unroll
    for (int m = 1; m < 16; ++m) mx = fmaxf(mx, msc[m]);
    float a[16], sum = 0.0f;
    #pragma unroll
    for (int m = 0; m < 16; ++m) { a[m] = __expf(msc[m] - mx); sum += a[m]; }
    const float inv = 1.0f / sum;
    #pragma unroll
    for (int m = 0; m < 16; ++m) a[m] *= inv;

    const float* cr  = cont + (long long)n * M_ * H_;
    float* orow = out + (long long)n * D_;

    for (int h = tid; h < H_; h += 256) {
        float s = 0.0f;
        #pragma unroll
        for (int m = 0; m < 16; ++m) s += a[m] * cr[m * H_ + h];
        orow[2 * H_ + h] = s;                     // weighted_sum block
    }
    if (tid < 2) orow[3 * H_ + tid] = ling[n * 2 + tid];
    if (tid < 20) {
        int w = width[n];
        w = (w < 0) ? 0 : ((w > 4) ? 4 : w);
        orow[3 * H_ + 2 + tid] = wtab[w * 20 + tid];
    }
}

// ---------------------------------------------------------------------------
extern "C" void kernel_launch(void* const* d_in, const int* in_sizes, int n_in,
                              void* d_out, int out_size, void* d_ws, size_t ws_size,
                              hipStream_t stream)
{
    const float* doc   = (const float*)d_in[0];   // (B,T,H)
    const float* smap  = (const float*)d_in[1];   // (B,S,T)
    const float* emap  = (const float*)d_in[2];   // (B,S,T)
    const float* cmap  = (const float*)d_in[3];   // (B,S,M,T)
    const float* ling  = (const float*)d_in[4];   // (B,S,2)
    const float* W1    = (const float*)d_in[5];   // (H,HID)
    const float* b1    = (const float*)d_in[6];   // (HID,)
    const float* W2    = (const float*)d_in[7];   // (HID,1)
    const float* b2    = (const float*)d_in[8];   // (1,)
    const float* wtab  = (const float*)d_in[9];   // (5,20)
    const int*   width = (const int*)d_in[10];    // (B,S)
    float* out = (float*)d_out;

    float* cont   = (float*)d_ws;                                 // B*S*M*H f32
    float* scores = cont + (size_t)B_ * S_ * M_ * H_;             // B*S*M f32

    // start -> out[:, 0:1024]
    se_wmma_gemm_f32<<<dim3(H_ / 32, S_ / 64, B_), 256, 0, stream>>>(
        smap, (long long)S_ * T_, T_,
        doc,  (long long)T_ * H_, H_,
        out,  (long long)S_ * D_, D_, T_);
    // end -> out[:, 1024:2048]
    se_wmma_gemm_f32<<<dim3(H_ / 32, S_ / 64, B_), 256, 0, stream>>>(
        emap, (long long)S_ * T_, T_,
        doc,  (long long)T_ * H_, H_,
        out + H_, (long long)S_ * D_, D_, T_);
    // cont -> workspace
    se_wmma_gemm_f32<<<dim3(H_ / 32, (S_ * M_) / 64, B_), 256, 0, stream>>>(
        cmap, (long long)S_ * M_ * T_, T_,
        doc,  (long long)T_ * H_, H_,
        cont, (long long)S_ * M_ * H_, H_, T_);

    se_score_kernel<<<B_ * S_, 128, 0, stream>>>(cont, W1, b1, W2, b2, scores);
    se_finalize_kernel<<<B_ * S_, 256, 0, stream>>>(cont, scores, ling, wtab, width, out);
}